// GNN_83356725280828
// MI455X (gfx1250) — compile-verified
//
#include <hip/hip_runtime.h>

typedef __attribute__((ext_vector_type(2))) float v2f;
typedef __attribute__((ext_vector_type(8))) float v8f;

#define DD 64   // feature dim

// ---------------- degree / normalization ----------------

__global__ void k_deg_init(float* __restrict__ deg, int n) {
    int i = blockIdx.x * blockDim.x + threadIdx.x;
    if (i < n) deg[i] = 1.0f;  // self-loop weight
}

__global__ void k_deg_scatter(const int* __restrict__ dst,
                              const float* __restrict__ ew,
                              float* __restrict__ deg, int nE) {
    int e = blockIdx.x * blockDim.x + threadIdx.x;
    if (e < nE) atomicAdd(&deg[dst[e]], ew[e]);
}

__global__ void k_dinv(float* __restrict__ deg, int n) {
    int i = blockIdx.x * blockDim.x + threadIdx.x;
    if (i < n) {
        float d = deg[i];
        deg[i] = (d > 0.0f) ? rsqrtf(d) : 0.0f;
    }
}

// norm[e] = dinv[src] * ew * dinv[dst]  (computed once, reused by both layers)
__global__ void k_edge_norm(const int* __restrict__ src,
                            const int* __restrict__ dst,
                            const float* __restrict__ ew,
                            const float* __restrict__ dinv,
                            float* __restrict__ norm, int nE) {
    int e = blockIdx.x * blockDim.x + threadIdx.x;
    if (e < nE) norm[e] = dinv[src[e]] * ew[e] * dinv[dst[e]];
}

// ---------------- fp32 WMMA GEMM: Y[N,64] = X[N,64] @ W[64,64] ----------------
// One wave32 computes one 16x16 output tile; K=64 via 16 chained 16x16x4 f32 WMMAs.
// A layout (ISA 7.12.2, 32-bit A 16x4): lanes 0-15 hold K={k,k+1}, lanes 16-31 K={k+2,k+3}.
// B/C/D: rows striped across lanes within a VGPR (lanes 16-31 carry the +half rows).
__global__ void k_gemm_wmma(const float* __restrict__ X,
                            const float* __restrict__ W,
                            float* __restrict__ Y, int nRows) {
    int wave = (blockIdx.x * blockDim.x + threadIdx.x) >> 5;
    int lane = threadIdx.x & 31;
    int mt = wave >> 2;          // 64/16 = 4 N-tiles
    int nt = wave & 3;
    if (mt * 16 >= nRows) return;          // wave-uniform: EXEC stays all-ones
    int row = mt * 16 + (lane & 15);       // A-matrix row for this lane
    int col = nt * 16 + (lane & 15);       // B/C/D column for this lane
    int khi = (lane >> 4) * 2;             // lanes 16-31 carry K+2 slots

    v8f c = {};
#pragma unroll
    for (int s = 0; s < 16; ++s) {
        int k = s * 4 + khi;
        v2f a, b;
        a.x = X[row * DD + k];
        a.y = X[row * DD + k + 1];
        b.x = W[k * DD + col];
        b.y = W[(k + 1) * DD + col];
        // (neg_a, A, neg_b, B, c_mod, C, reuse_a, reuse_b)
        c = __builtin_amdgcn_wmma_f32_16x16x4_f32(false, a, false, b,
                                                  (short)0, c, false, false);
    }
    int mout = mt * 16 + (lane >> 4) * 8;  // C/D: VGPR v -> row v (lanes 0-15) / v+8 (16-31)
#pragma unroll
    for (int v = 0; v < 8; ++v)
        Y[(mout + v) * DD + col] = c[v];
}

// ---------------- GCN aggregation ----------------

// agg[i,:] = hW[i,:] * dinv[i]^2   (self-loop contribution), float4-vectorized
__global__ void k_self_init4(const float4* __restrict__ hW,
                             const float* __restrict__ dinv,
                             float4* __restrict__ agg, int nd4) {
    int t = blockIdx.x * blockDim.x + threadIdx.x;
    if (t < nd4) {
        float di = dinv[t >> 4];   // 16 float4 per node
        di *= di;
        float4 v = hW[t];
        v.x *= di; v.y *= di; v.z *= di; v.w *= di;
        agg[t] = v;
    }
}

// 16 lanes per edge, float4 gather + 4 atomic adds per lane:
// agg[dst,:] += hW[src,:] * norm[e]
__global__ void k_edge_scatter(const float* __restrict__ hW,
                               const int* __restrict__ src,
                               const int* __restrict__ dst,
                               const float* __restrict__ norm,
                               float* __restrict__ agg, int nE) {
    int t = blockIdx.x * blockDim.x + threadIdx.x;
    int e = t >> 4;
    if (e >= nE) return;
    int d4 = (t & 15) << 2;
    int s = src[e];
    int q = dst[e];
    float nm = norm[e];
    float4 v = *reinterpret_cast<const float4*>(hW + s * DD + d4);  // global_load_b128
    float* ap = agg + q * DD + d4;
    atomicAdd(ap + 0, v.x * nm);
    atomicAdd(ap + 1, v.y * nm);
    atomicAdd(ap + 2, v.z * nm);
    atomicAdd(ap + 3, v.w * nm);
}

// h = relu(agg + b), in place, float4-vectorized
__global__ void k_bias_relu4(float4* __restrict__ h,
                             const float4* __restrict__ b, int nd4) {
    int t = blockIdx.x * blockDim.x + threadIdx.x;
    if (t < nd4) {
        float4 v = h[t];
        float4 bb = b[t & 15];
        v.x = fmaxf(v.x + bb.x, 0.0f);
        v.y = fmaxf(v.y + bb.y, 0.0f);
        v.z = fmaxf(v.z + bb.z, 0.0f);
        v.w = fmaxf(v.w + bb.w, 0.0f);
        h[t] = v;
    }
}

// ---------------- global mean pool ----------------

__global__ void k_pool_zero(float* __restrict__ p, int n) {
    int i = blockIdx.x * blockDim.x + threadIdx.x;
    if (i < n) p[i] = 0.0f;
}

__global__ void k_pool_scatter(const float* __restrict__ h,
                               const int* __restrict__ batch,
                               float* __restrict__ pool,
                               float* __restrict__ cnt, int nd) {
    int t = blockIdx.x * blockDim.x + threadIdx.x;
    if (t >= nd) return;
    int node = t >> 6;
    int d = t & 63;
    int g = batch[node];
    atomicAdd(&pool[g * DD + d], h[t]);
    if (d == 0) atomicAdd(&cnt[g], 1.0f);
}

__global__ void k_pool_final(const float* __restrict__ pool,
                             const float* __restrict__ cnt,
                             float* __restrict__ out, int gd) {
    int i = blockIdx.x * blockDim.x + threadIdx.x;
    if (i < gd) out[i] = pool[i] / fmaxf(cnt[i >> 6], 1.0f);
}

// ---------------- launch ----------------

extern "C" void kernel_launch(void* const* d_in, const int* in_sizes, int n_in,
                              void* d_out, int out_size, void* d_ws, size_t ws_size,
                              hipStream_t stream) {
    const float* x     = (const float*)d_in[0];
    const int*   ei    = (const int*)d_in[1];
    const int*   batch = (const int*)d_in[2];
    const float* ew    = (const float*)d_in[3];
    const float* W1    = (const float*)d_in[4];
    const float* b1    = (const float*)d_in[5];
    const float* W2    = (const float*)d_in[6];
    const float* b2    = (const float*)d_in[7];
    float* out = (float*)d_out;

    const int N = in_sizes[0] / DD;       // 100000 (divisible by 16)
    const int E = in_sizes[3];            // 1600000
    const int G = out_size / DD;          // 64
    const int* src = ei;
    const int* dst = ei + E;

    // workspace layout (floats): dinv | norm | bufA | bufB | pool | cnt  (~58 MB)
    float* ws = (float*)d_ws;
    size_t nAl = ((size_t)N + 255) & ~(size_t)255;
    size_t eAl = ((size_t)E + 255) & ~(size_t)255;
    float* dinv = ws;
    float* norm = ws + nAl;
    float* A    = norm + eAl;
    float* B    = A + (size_t)N * DD;
    float* pool = B + (size_t)N * DD;
    float* cnt  = pool + (size_t)G * DD;

    const int ND  = N * DD;
    const int ND4 = ND / 4;
    const int TB  = 256;

    // GCN normalization: deg = 1 + sum(ew @ dst); dinv = rsqrt(deg); edge norms
    k_deg_init<<<(N + TB - 1) / TB, TB, 0, stream>>>(dinv, N);
    k_deg_scatter<<<(E + TB - 1) / TB, TB, 0, stream>>>(dst, ew, dinv, E);
    k_dinv<<<(N + TB - 1) / TB, TB, 0, stream>>>(dinv, N);
    k_edge_norm<<<(E + TB - 1) / TB, TB, 0, stream>>>(src, dst, ew, dinv, norm, E);

    int mTiles = (N + 15) / 16;
    int gemmBlocks = (mTiles * 4 * 32 + TB - 1) / TB;   // 4 N-tiles, 32 lanes/wave
    int scatBlocks = (E * 16 + TB - 1) / TB;            // 16 lanes/edge

    // layer 1: A = x@W1; B = agg; B = relu(B + b1)
    k_gemm_wmma<<<gemmBlocks, TB, 0, stream>>>(x, W1, A, N);
    k_self_init4<<<(ND4 + TB - 1) / TB, TB, 0, stream>>>((const float4*)A, dinv, (float4*)B, ND4);
    k_edge_scatter<<<scatBlocks, TB, 0, stream>>>(A, src, dst, norm, B, E);
    k_bias_relu4<<<(ND4 + TB - 1) / TB, TB, 0, stream>>>((float4*)B, (const float4*)b1, ND4);

    // layer 2 (ping-pong: gemm consumes B into A, then B is overwritten)
    k_gemm_wmma<<<gemmBlocks, TB, 0, stream>>>(B, W2, A, N);
    k_self_init4<<<(ND4 + TB - 1) / TB, TB, 0, stream>>>((const float4*)A, dinv, (float4*)B, ND4);
    k_edge_scatter<<<scatBlocks, TB, 0, stream>>>(A, src, dst, norm, B, E);
    k_bias_relu4<<<(ND4 + TB - 1) / TB, TB, 0, stream>>>((float4*)B, (const float4*)b2, ND4);

    // mean pool: pool/cnt contiguous, zero together
    k_pool_zero<<<(G * DD + G + TB - 1) / TB, TB, 0, stream>>>(pool, G * DD + G);
    k_pool_scatter<<<(ND + TB - 1) / TB, TB, 0, stream>>>(B, batch, pool, cnt, ND);
    k_pool_final<<<(G * DD + TB - 1) / TB, TB, 0, stream>>>(pool, cnt, out, G * DD);
}